// RN_L_15470472200842
// MI455X (gfx1250) — compile-verified
//
#include <hip/hip_runtime.h>
#include <math.h>

#define BB   8
#define CC   64
#define MM   256
#define NNN  256
#define MN   (MM * NNN)        // 65536
#define PIX  (BB * MN)         // 524288 pixels per map
#define NTOTF 524288.0f
#define THRESH 0.8f
#define EPSV 1e-5f

#define HALO_W 136             // 128 tile cols + 6 halo + pad (also avoids bank issues)
#define CH_STRIDE (8 * HALO_W) // 8 rows (7 used + 1 zero row) = 1088 floats/channel

typedef float v2f __attribute__((ext_vector_type(2)));
typedef float v8f __attribute__((ext_vector_type(8)));

// ---------------------------------------------------------------------------
// Kernel 1: per-pixel mean and max over the 64 channels (float4 vectorized).
// ---------------------------------------------------------------------------
__global__ __launch_bounds__(256) void k_chanreduce(const float* __restrict__ x,
                                                    float* __restrict__ avgm,
                                                    float* __restrict__ maxm) {
    int t   = blockIdx.x * 256 + threadIdx.x;        // PIX/4 threads total
    int b   = t / (MN / 4);
    int sp4 = (t - b * (MN / 4)) * 4;
    const float4* xb = (const float4*)(x + (size_t)b * CC * MN + sp4);
    float4 s = make_float4(0.f, 0.f, 0.f, 0.f);
    float4 m = make_float4(-3.4e38f, -3.4e38f, -3.4e38f, -3.4e38f);
#pragma unroll 8
    for (int c = 0; c < CC; ++c) {
        float4 v = xb[c * (MN / 4)];
        s.x += v.x; s.y += v.y; s.z += v.z; s.w += v.w;
        m.x = fmaxf(m.x, v.x); m.y = fmaxf(m.y, v.y);
        m.z = fmaxf(m.z, v.z); m.w = fmaxf(m.w, v.w);
    }
    const float inv = 1.0f / 64.0f;
    float4 a = make_float4(s.x * inv, s.y * inv, s.z * inv, s.w * inv);
    int p = b * MN + sp4;
    *(float4*)(avgm + p) = a;
    *(float4*)(maxm + p) = m;
}

// ---------------------------------------------------------------------------
// Kernel 2: 7x7 conv (2ch -> 1) + sigmoid via V_WMMA_F32_16X16X4_F32.
// Block = 128 consecutive pixels of one image row; zero-padded halo staged in
// LDS (2ch x 8rows x 136cols), weights repacked K = ch*64 + r*8 + cc (pad=0).
// Inner loop is branch-free: bit-op index decode, LDS reads, WMMA accumulate.
// A layout (16x4 f32): lanes 0-15 -> K=0(x),1(y); lanes 16-31 -> K=2(x),3(y).
// D layout: N=0 column lives in lanes 0 (M=0..7) and 16 (M=8..15).
// ---------------------------------------------------------------------------
__global__ __launch_bounds__(256) void k_conv1_wmma(const float* __restrict__ avgm,
                                                    const float* __restrict__ maxm,
                                                    const float* __restrict__ w1,
                                                    float* __restrict__ sam) {
    __shared__ float ldsA[2 * CH_STRIDE];   // 2176 floats
    __shared__ float ldsW[128];
    int tid = threadIdx.x;
    int blockbase = blockIdx.x * 128;
    int b  = blockbase >> 16;
    int i  = (blockbase & (MN - 1)) >> 8;
    int j0 = blockbase & 255;               // 0 or 128

    // ---- stage halo (one bounds check per element, zero fill) ----
    for (int e = tid; e < 2 * CH_STRIDE; e += 256) {
        int ch  = e / CH_STRIDE;
        int rem = e - ch * CH_STRIDE;
        int rr  = rem / HALO_W;
        int col = rem - rr * HALO_W;
        float v = 0.f;
        int ii = i + rr - 3;
        int jj = j0 + col - 3;
        if (rr < 7 && (unsigned)ii < 256u && (unsigned)jj < 256u)
            v = (ch ? maxm : avgm)[(b << 16) + (ii << 8) + jj];
        ldsA[e] = v;
    }
    if (tid < 128) {
        int ch = tid >> 6, r = (tid >> 3) & 7, cc = tid & 7;
        ldsW[tid] = (r < 7 && cc < 7) ? w1[ch * 49 + r * 7 + cc] : 0.f;
    }
    __syncthreads();

    int lane   = tid & 31;
    int m      = lane & 15;
    int half   = lane >> 4;
    int waveId = tid >> 5;
    int jl     = waveId * 16 + m;           // 0..127 local pixel column
    float c0   = (m == 0) ? 1.f : 0.f;

    v8f acc = {0.f, 0.f, 0.f, 0.f, 0.f, 0.f, 0.f, 0.f};
#pragma unroll
    for (int t = 0; t < 32; ++t) {          // K = 128 (padded)
        int ka = t * 4 + half * 2;          // even -> cc(ka) <= 6, kb same row
        int ch = ka >> 6;
        int r  = (ka >> 3) & 7;
        int cc = ka & 7;
        int addr = ch * CH_STRIDE + r * HALO_W + jl + cc;
        v2f a, w;
        a.x = ldsA[addr];
        a.y = ldsA[addr + 1];
        w.x = ldsW[ka] * c0;
        w.y = ldsW[ka + 1] * c0;
        acc = __builtin_amdgcn_wmma_f32_16x16x4_f32(false, a, false, w,
                                                    (short)0, acc, false, false);
    }
    int p0 = blockbase + waveId * 16;
    if (m == 0) {
#pragma unroll
        for (int r = 0; r < 8; ++r) {
            float v  = acc[r];
            float sg = 1.0f / (1.0f + __expf(-v));
            sam[p0 + half * 8 + r] = sg;
        }
    }
}

// ---------------------------------------------------------------------------
// Kernel 3: gamma & beta 7x7 convs in ONE WMMA pass: B column 0 = wg,
// column 1 = wb. K padded to 64 (r*8 + cc) -> 16 branch-free WMMA steps.
// ---------------------------------------------------------------------------
__global__ __launch_bounds__(256) void k_convgb_wmma(const float* __restrict__ sam,
                                                     const float* __restrict__ wg,
                                                     const float* __restrict__ bgp,
                                                     const float* __restrict__ wb,
                                                     const float* __restrict__ bbp,
                                                     float* __restrict__ gam,
                                                     float* __restrict__ bet) {
    __shared__ float ldsA[CH_STRIDE];       // 1088 floats
    __shared__ float ldsWg[64];
    __shared__ float ldsWb[64];
    int tid = threadIdx.x;
    int blockbase = blockIdx.x * 128;
    int b  = blockbase >> 16;
    int i  = (blockbase & (MN - 1)) >> 8;
    int j0 = blockbase & 255;

    for (int e = tid; e < CH_STRIDE; e += 256) {
        int rr  = e / HALO_W;
        int col = e - rr * HALO_W;
        float v = 0.f;
        int ii = i + rr - 3;
        int jj = j0 + col - 3;
        if (rr < 7 && (unsigned)ii < 256u && (unsigned)jj < 256u)
            v = sam[(b << 16) + (ii << 8) + jj];
        ldsA[e] = v;
    }
    if (tid < 64) {
        int r = tid >> 3, cc = tid & 7;
        float vg = 0.f, vb = 0.f;
        if (r < 7 && cc < 7) { vg = wg[r * 7 + cc]; vb = wb[r * 7 + cc]; }
        ldsWg[tid] = vg;
        ldsWb[tid] = vb;
    }
    __syncthreads();

    int lane   = tid & 31;
    int m      = lane & 15;
    int half   = lane >> 4;
    int waveId = tid >> 5;
    int jl     = waveId * 16 + m;
    float c0   = (m == 0) ? 1.f : 0.f;      // gamma column (N=0)
    float c1   = (m == 1) ? 1.f : 0.f;      // beta column  (N=1)

    v8f acc = {0.f, 0.f, 0.f, 0.f, 0.f, 0.f, 0.f, 0.f};
#pragma unroll
    for (int t = 0; t < 16; ++t) {          // K = 64 (padded)
        int ka = t * 4 + half * 2;
        int r  = (ka >> 3) & 7;
        int cc = ka & 7;
        int addr = r * HALO_W + jl + cc;
        v2f a, w;
        a.x = ldsA[addr];
        a.y = ldsA[addr + 1];
        w.x = ldsWg[ka] * c0 + ldsWb[ka] * c1;
        w.y = ldsWg[ka + 1] * c0 + ldsWb[ka + 1] * c1;
        acc = __builtin_amdgcn_wmma_f32_16x16x4_f32(false, a, false, w,
                                                    (short)0, acc, false, false);
    }
    int p0 = blockbase + waveId * 16;
    if (m == 0) {                           // N=0 column -> gamma
        float bgv = bgp[0];
#pragma unroll
        for (int r = 0; r < 8; ++r) gam[p0 + half * 8 + r] = acc[r] + bgv;
    } else if (m == 1) {                    // N=1 column -> beta
        float bbv = bbp[0];
#pragma unroll
        for (int r = 0; r < 8; ++r) bet[p0 + half * 8 + r] = acc[r] + bbv;
    }
}

// ---------------------------------------------------------------------------
// Kernel 4a: per-(channel, chunk) partial sums: Σx, Σx², Σmx, Σmx², Σm.
// Deterministic (no atomics): each block writes its own partial slots.
// Grid: 64 channels * 256 chunks; each block covers 2048 pixels.
// ---------------------------------------------------------------------------
__global__ __launch_bounds__(256) void k_stats_part(const float* __restrict__ x,
                                                    const float* __restrict__ sam,
                                                    float* __restrict__ parts) {
    int c     = blockIdx.x >> 8;
    int chunk = blockIdx.x & 255;
    int tid   = threadIdx.x;
    float sx = 0.f, sx2 = 0.f, smx = 0.f, smx2 = 0.f, cnt = 0.f;
#pragma unroll
    for (int k = 0; k < 2; ++k) {
        int p  = chunk * 2048 + k * 1024 + tid * 4;
        int b  = p >> 16;
        int sp = p & (MN - 1);
        float4 xv = *(const float4*)(x + ((size_t)(b * CC + c) << 16) + sp);
        float4 sv = *(const float4*)(sam + p);
        float mk;
        mk = (sv.x >= THRESH) ? 1.f : 0.f; sx += xv.x; sx2 += xv.x * xv.x; smx += mk * xv.x; smx2 += mk * xv.x * xv.x; cnt += mk;
        mk = (sv.y >= THRESH) ? 1.f : 0.f; sx += xv.y; sx2 += xv.y * xv.y; smx += mk * xv.y; smx2 += mk * xv.y * xv.y; cnt += mk;
        mk = (sv.z >= THRESH) ? 1.f : 0.f; sx += xv.z; sx2 += xv.z * xv.z; smx += mk * xv.z; smx2 += mk * xv.z * xv.z; cnt += mk;
        mk = (sv.w >= THRESH) ? 1.f : 0.f; sx += xv.w; sx2 += xv.w * xv.w; smx += mk * xv.w; smx2 += mk * xv.w * xv.w; cnt += mk;
    }
    __shared__ float red[5][256];
    red[0][tid] = sx; red[1][tid] = sx2; red[2][tid] = smx;
    red[3][tid] = smx2; red[4][tid] = cnt;
    __syncthreads();
    for (int s = 128; s > 0; s >>= 1) {
        if (tid < s) {
            red[0][tid] += red[0][tid + s];
            red[1][tid] += red[1][tid + s];
            red[2][tid] += red[2][tid + s];
            red[3][tid] += red[3][tid + s];
            red[4][tid] += red[4][tid + s];
        }
        __syncthreads();
    }
    if (tid == 0) {
#pragma unroll
        for (int q = 0; q < 5; ++q) parts[q * (64 * 256) + blockIdx.x] = red[q][0];
    }
}

// ---------------------------------------------------------------------------
// Kernel 4b: reduce 256 partials per channel (fixed tree -> deterministic)
// and finalize {mu1, f1, mu0, f0} per channel (closed-form region_norm).
// ---------------------------------------------------------------------------
__global__ __launch_bounds__(256) void k_stats_final(const float* __restrict__ parts,
                                                     float* __restrict__ params) {
    int c = blockIdx.x, tid = threadIdx.x;
    __shared__ float red[5][256];
#pragma unroll
    for (int q = 0; q < 5; ++q) red[q][tid] = parts[q * (64 * 256) + c * 256 + tid];
    __syncthreads();
    for (int s = 128; s > 0; s >>= 1) {
        if (tid < s) {
            red[0][tid] += red[0][tid + s];
            red[1][tid] += red[1][tid + s];
            red[2][tid] += red[2][tid + s];
            red[3][tid] += red[3][tid + s];
            red[4][tid] += red[4][tid + s];
        }
        __syncthreads();
    }
    if (tid == 0) {
        float s_x  = red[0][0], s_x2 = red[1][0];
        float s_mx = red[2][0], s_mx2 = red[3][0];
        float Sr   = red[4][0];
        float S0   = NTOTF - Sr;
        float Srd  = (Sr == 0.f) ? 1.f : Sr;
        float S0d  = (S0 == 0.f) ? 1.f : S0;
        float mu1  = s_mx / Srd;
        float v1   = fmaxf((s_mx2 - s_mx * mu1) / NTOTF, 0.f);
        float f1   = rsqrtf(v1 + EPSV) * sqrtf(Sr / NTOTF);
        float s0   = s_x - s_mx, ss0 = s_x2 - s_mx2;
        float mu0  = s0 / S0d;
        float v0   = fmaxf((ss0 - s0 * mu0) / NTOTF, 0.f);
        float f0   = rsqrtf(v0 + EPSV) * sqrtf(S0 / NTOTF);
        params[c * 4 + 0] = mu1;
        params[c * 4 + 1] = f1;
        params[c * 4 + 2] = mu0;
        params[c * 4 + 3] = f0;
    }
}

// ---------------------------------------------------------------------------
// Kernel 5: out = (mask ? (x-mu1)*f1 : (x-mu0)*f0) * (1+gamma) + beta
// float4 streaming; each block stays within one (b,c) plane -> uniform params.
// ---------------------------------------------------------------------------
__global__ __launch_bounds__(256) void k_apply(const float* __restrict__ x,
                                               const float* __restrict__ sam,
                                               const float* __restrict__ gam,
                                               const float* __restrict__ bet,
                                               const float* __restrict__ params,
                                               float* __restrict__ out) {
    int g = blockIdx.x * 256 + threadIdx.x;
    size_t e = (size_t)g * 4;
    int b  = (int)(e >> 22);                 // C*MN = 2^22
    int r  = (int)(e & ((1u << 22) - 1));
    int c  = r >> 16;
    int sp = r & (MN - 1);
    int p  = (b << 16) + sp;
    float mu1 = params[c * 4 + 0], f1 = params[c * 4 + 1];
    float mu0 = params[c * 4 + 2], f0 = params[c * 4 + 3];
    float4 xv = *(const float4*)(x + e);
    float4 sv = *(const float4*)(sam + p);
    float4 gv = *(const float4*)(gam + p);
    float4 bv = *(const float4*)(bet + p);
    float4 o;
    o.x = ((sv.x >= THRESH) ? (xv.x - mu1) * f1 : (xv.x - mu0) * f0) * (1.f + gv.x) + bv.x;
    o.y = ((sv.y >= THRESH) ? (xv.y - mu1) * f1 : (xv.y - mu0) * f0) * (1.f + gv.y) + bv.y;
    o.z = ((sv.z >= THRESH) ? (xv.z - mu1) * f1 : (xv.z - mu0) * f0) * (1.f + gv.z) + bv.z;
    o.w = ((sv.w >= THRESH) ? (xv.w - mu1) * f1 : (xv.w - mu0) * f0) * (1.f + gv.w) + bv.w;
    *(float4*)(out + e) = o;
}

// ---------------------------------------------------------------------------
extern "C" void kernel_launch(void* const* d_in, const int* in_sizes, int n_in,
                              void* d_out, int out_size, void* d_ws, size_t ws_size,
                              hipStream_t stream) {
    const float* x   = (const float*)d_in[0];
    const float* w1  = (const float*)d_in[1];   // 98
    const float* wg  = (const float*)d_in[2];   // 49
    const float* bgp = (const float*)d_in[3];   // 1
    const float* wb  = (const float*)d_in[4];   // 49
    const float* bbp = (const float*)d_in[5];   // 1
    float* out = (float*)d_out;

    float* ws     = (float*)d_ws;
    float* avgm   = ws;                  // PIX
    float* maxm   = ws + (size_t)PIX;    // PIX
    float* sam    = ws + (size_t)2 * PIX;
    float* gam    = ws + (size_t)3 * PIX;
    float* bet    = ws + (size_t)4 * PIX;
    float* parts  = ws + (size_t)5 * PIX;          // 5 * 64 * 256
    float* params = parts + 5 * 64 * 256;          // 64 * 4

    k_chanreduce<<<PIX / 4 / 256, 256, 0, stream>>>(x, avgm, maxm);
    k_conv1_wmma<<<PIX / 128, 256, 0, stream>>>(avgm, maxm, w1, sam);
    k_convgb_wmma<<<PIX / 128, 256, 0, stream>>>(sam, wg, bgp, wb, bbp, gam, bet);
    k_stats_part<<<64 * 256, 256, 0, stream>>>(x, sam, parts);
    k_stats_final<<<64, 256, 0, stream>>>(parts, params);
    k_apply<<<(BB * CC * MN / 4) / 256, 256, 0, stream>>>(x, sam, gam, bet, params, out);
}